// FreqActivation_22187801051333
// MI455X (gfx1250) — compile-verified
//
#include <hip/hip_runtime.h>
#include <math.h>

// FreqActivation for MI455X (gfx1250).
// out[b,h,w,c] (complex64 as interleaved float pairs):
//   top region  (h==0 || w==0 || h+w<=256): F(h,w)
//   bot region  (h,w>=1 && h+w>=257):       conj(F(256-h,256-w))
// Fused mapping: each "active" (top-region) pixel loads its input once and
// writes its own output; strict-interior top pixels (h,w>=1, h+w<=255)
// additionally write the conjugate to the mirrored bottom pixel.
// -> reads ~67MB, writes 134MB: ~201MB HBM traffic, ~8.6us at 23.3 TB/s.
//
// CDNA5 paths: per-lane async copy global->LDS (ASYNCcnt) via
// global_load_async_to_lds_b128 + s_wait_asynccnt + ds_load_b128, and the
// native V_TANH_F32 / V_SIN_F32 / V_COS_F32 transcendentals.
// Math: ph = pi*tanh(p); hw sin/cos take revolutions (x/2pi), so
//   sin(ph) = v_sin(tanh(p)/2), cos(ph) = v_cos(tanh(p)/2).

typedef float v4f __attribute__((ext_vector_type(4)));

#define BLOCK 256
#define WAVES 8  // BLOCK / 32

__device__ __forceinline__ float hw_tanh(float x) {
#if __has_builtin(__builtin_amdgcn_tanhf)
  return __builtin_amdgcn_tanhf(x);
#else
  float r;
  asm("v_tanh_f32 %0, %1" : "=v"(r) : "v"(x));
  return r;
#endif
}

// sin(2*pi*u) / cos(2*pi*u): hardware trans ops take pre-normalized input.
__device__ __forceinline__ float hw_sin_rev(float u) { return __builtin_amdgcn_sinf(u); }
__device__ __forceinline__ float hw_cos_rev(float u) { return __builtin_amdgcn_cosf(u); }

__device__ __forceinline__ void async_load16(unsigned lds_off, const void* gptr) {
  // GV mode: vdst = VGPR holding LDS byte offset, vaddr = 64-bit global addr.
  asm volatile("global_load_async_to_lds_b128 %0, %1, off"
               :
               : "v"(lds_off), "v"((unsigned long long)(size_t)gptr)
               : "memory");
}

__global__ __launch_bounds__(BLOCK) void
freq_activation_kernel(const float* __restrict__ in, float* __restrict__ out,
                       int niter, unsigned stride) {
  __shared__ v4f smem[WAVES][32];
  const unsigned lane = threadIdx.x & 31u;
  const unsigned wid  = threadIdx.x >> 5;
  // Low 32 bits of the generic pointer to shared == LDS byte offset.
  const unsigned lds_off = (unsigned)(size_t)(&smem[wid][lane]);

  unsigned t = blockIdx.x * BLOCK + threadIdx.x;
  for (int k = 0; k < niter; ++k, t += stride) {
    const unsigned c4  = t & 15u;         // channel pair 0..15 (2 channels each)
    const unsigned pix = t >> 4;          // linear (b,h,w)
    const unsigned w   = pix & 255u;
    const unsigned h   = (pix >> 8) & 255u;
    const bool active  = (h == 0u) || (w == 0u) || ((h + w) <= 256u);

    if (active) {
      // input float offset: pix*64 + c4*4  ([B,H,W,C,2] layout, C=32)
      async_load16(lds_off, in + ((unsigned long long)pix * 64ull + (unsigned long long)c4 * 4u));
    }
    asm volatile("s_wait_asynccnt 0" ::: "memory");

    if (active) {
      v4f v = smem[wid][lane];  // {amp_raw0, ph_raw0, amp_raw1, ph_raw1}

      float amp0 = fmaxf(v.x, 0.0f);
      float amp1 = fmaxf(v.z, 0.0f);
      // u = tanh(p)/2 revolutions  <=>  ph = pi*tanh(p) radians
      float u0 = 0.5f * hw_tanh(v.y);
      float u1 = 0.5f * hw_tanh(v.w);

      v4f o;
      o.x = amp0 * hw_cos_rev(u0);  // re0
      o.y = amp0 * hw_sin_rev(u0);  // im0
      o.z = amp1 * hw_cos_rev(u1);  // re1
      o.w = amp1 * hw_sin_rev(u1);  // im1

      v4f* dst = (v4f*)(out + ((unsigned long long)pix * 64ull + (unsigned long long)c4 * 4u));
      __builtin_nontemporal_store(o, dst);

      // Strict interior of the top triangle also owns the mirrored bottom output.
      if ((h >= 1u) && (w >= 1u) && ((h + w) <= 255u)) {
        unsigned pixm = (pix & 0xFFFF0000u) | ((256u - h) << 8) | (256u - w);
        v4f om;
        om.x =  o.x;
        om.y = -o.y;
        om.z =  o.z;
        om.w = -o.w;
        v4f* dstm = (v4f*)(out + ((unsigned long long)pixm * 64ull + (unsigned long long)c4 * 4u));
        __builtin_nontemporal_store(om, dstm);
      }
    }
  }
}

extern "C" void kernel_launch(void* const* d_in, const int* in_sizes, int n_in,
                              void* d_out, int out_size, void* d_ws, size_t ws_size,
                              hipStream_t stream) {
  (void)in_sizes; (void)n_in; (void)out_size; (void)d_ws; (void)ws_size;
  const float* in = (const float*)d_in[0];
  float* out = (float*)d_out;

  // B=8, H=256, W=256, C=32; one thread per (pixel, channel-pair)
  const unsigned total_threads = 8u * 256u * 256u * 16u;  // 8,388,608
  const int niter = 4;
  const unsigned stride = total_threads / (unsigned)niter; // 2,097,152
  const unsigned grid = stride / BLOCK;                    // 8192 blocks

  freq_activation_kernel<<<grid, BLOCK, 0, stream>>>(in, out, niter, stride);
}